// SpectralSubtraction_64175401337096
// MI455X (gfx1250) — compile-verified
//
#include <hip/hip_runtime.h>

// Spectral subtraction: x (B,2,R,C) f32 -> out (B,2,R,C) f32
// Per row: noise = mean of n_avg smallest squares of mag; m = relu(mag - noise);
// out = [m*cos(phase), m*sin(phase)].
//
// Memory-bound (~537 MB @ 23.3 TB/s ~ 23 us). One 256-thread block (8 wave32)
// per row. Mag row staged via CDNA5 GLOBAL_LOAD_ASYNC_TO_LDS_B128 (ASYNCcnt);
// exact k-smallest via 4-pass 8-bit radix select:
//   pass 0: per-wave privatized LDS histograms (bank-skewed) -> low contention
//   scan/select: single wave32 (__shfl_up) -> 3 block barriers per pass.

#define COLS 2048
#define TPB  256
#define VPT  (COLS / TPB)   // 8 floats per thread
#define NW   (TPB / 32)     // 8 waves
#define HPAD 8              // skew: copy w, bin b -> bank (b + 8w) & 63

__global__ __launch_bounds__(TPB) void spectral_sub_kernel(
    const float* __restrict__ x, const int* __restrict__ n_avg_ptr,
    float* __restrict__ out, int nrow, int chan_stride)
{
    __shared__ __align__(16) float smag[COLS];
    __shared__ int   histp[NW][TPB + HPAD];   // pass-0 privatized histograms
    __shared__ int   hist[TPB];               // passes 1..3 (few actives)
    __shared__ int   s_byte, s_krem;
    __shared__ float s_sum;

    const int t   = threadIdx.x;
    const int wv  = t >> 5;
    const int rid = blockIdx.x;               // 0 .. B*nrow-1
    const int b   = rid / nrow;
    const int r   = rid - b * nrow;

    const float* magp = x + (size_t)b * 2u * (size_t)chan_stride + (size_t)r * COLS;
    const float* php  = magp + chan_stride;

    // ---- stage magnitude row into LDS with the async data mover ----
    {
        unsigned lds0 = (unsigned)(size_t)&smag[t * VPT];
        unsigned voff = (unsigned)(t * VPT * sizeof(float));
        unsigned long long gbase = (unsigned long long)magp;
        asm volatile(
            "global_load_async_to_lds_b128 %0, %1, %2\n\t"
            "global_load_async_to_lds_b128 %0, %1, %2 offset:16"
            :
            : "v"(lds0), "v"(voff), "s"(gbase)
            : "memory");
    }

    // Phase loads issued now so they fly while we wait on the async copy.
    const float4* ph4 = (const float4*)(php + t * VPT);
    float4 p0 = ph4[0];
    float4 p1 = ph4[1];

    asm volatile("s_wait_asynccnt 0" ::: "memory");
    __syncthreads();

    // ---- my 8 mags from LDS; squares as uint bits (monotone for x >= 0) ----
    const float4* m4 = (const float4*)&smag[t * VPT];
    float4 m0 = m4[0];
    float4 m1 = m4[1];
    float mag[VPT] = {m0.x, m0.y, m0.z, m0.w, m1.x, m1.y, m1.z, m1.w};

    unsigned sq[VPT];
#pragma unroll
    for (int j = 0; j < VPT; ++j) {
        float s2 = mag[j] * mag[j];
        sq[j] = __float_as_uint(s2);
    }

    const int navg = n_avg_ptr[0];   // 32
    int k = navg;
    unsigned prefix = 0u;

    // ================= pass 0 (shift 24): privatized histograms =============
#pragma unroll
    for (int w = 0; w < NW; ++w) histp[w][t] = 0;
    __syncthreads();
#pragma unroll
    for (int j = 0; j < VPT; ++j) {
        atomicAdd(&histp[wv][sq[j] >> 24], 1);     // ds_add, contention / NW
    }
    __syncthreads();
    if (t < 32) {                                  // wave 0: merge + scan + pick
        const int base = t * 8;
        int p[8];
        int run = 0;
#pragma unroll
        for (int j = 0; j < 8; ++j) {
            int c = 0;
#pragma unroll
            for (int w = 0; w < NW; ++w) c += histp[w][base + j];
            run += c;
            p[j] = run;
        }
        int incl = run;                            // wave32 inclusive scan
#pragma unroll
        for (int d = 1; d < 32; d <<= 1) {
            int up = __shfl_up(incl, d, 32);
            if (t >= d) incl += up;
        }
        const int excl = incl - run;
#pragma unroll
        for (int j = 0; j < 8; ++j) {
            int cb = excl + (j ? p[j - 1] : 0);
            int ci = excl + p[j];
            if (cb < k && k <= ci) { s_byte = base + j; s_krem = k - cb; }
        }
    }
    __syncthreads();
    prefix = ((unsigned)s_byte) << 24;
    k = s_krem;

    // ================= passes 1..3: few active elements =====================
    for (int shift = 16; shift >= 0; shift -= 8) {
        hist[t] = 0;
        __syncthreads();
        const unsigned himask = 0xFFFFFFFFu << (shift + 8);
#pragma unroll
        for (int j = 0; j < VPT; ++j) {
            if (((sq[j] ^ prefix) & himask) == 0u) {
                atomicAdd(&hist[(sq[j] >> shift) & 0xFFu], 1);
            }
        }
        __syncthreads();
        if (t < 32) {
            const int base = t * 8;
            int p[8];
            int run = 0;
#pragma unroll
            for (int j = 0; j < 8; ++j) {
                int c = hist[base + j];
                run += c;
                p[j] = run;
            }
            int incl = run;
#pragma unroll
            for (int d = 1; d < 32; d <<= 1) {
                int up = __shfl_up(incl, d, 32);
                if (t >= d) incl += up;
            }
            const int excl = incl - run;
#pragma unroll
            for (int j = 0; j < 8; ++j) {
                int cb = excl + (j ? p[j - 1] : 0);
                int ci = excl + p[j];
                if (cb < k && k <= ci) { s_byte = base + j; s_krem = k - cb; }
            }
        }
        __syncthreads();
        prefix |= ((unsigned)s_byte) << shift;
        k = s_krem;
    }

    const unsigned T  = prefix;               // bits of k-th smallest square
    const float    Tf = __uint_as_float(T);

    // ---- noise = (sum of squares strictly < T + k_rem * T) / n_avg ----
    if (t == 0) s_sum = 0.0f;
    __syncthreads();
    float part = 0.0f;
#pragma unroll
    for (int j = 0; j < VPT; ++j) {
        if (sq[j] < T) part += __uint_as_float(sq[j]);
    }
    if (part != 0.0f) atomicAdd(&s_sum, part);   // ds_add_f32, <=32 hits/row
    __syncthreads();

    const float noise = (s_sum + (float)k * Tf) / (float)navg;

    // ---- m = relu(mag - noise); out = [m*cos(ph), m*sin(ph)] ----
    float ph[VPT] = {p0.x, p0.y, p0.z, p0.w, p1.x, p1.y, p1.z, p1.w};
    float outr[VPT], outi[VPT];
#pragma unroll
    for (int j = 0; j < VPT; ++j) {
        float m = fmaxf(mag[j] - noise, 0.0f);
        float sn, cs;
        __sincosf(ph[j], &sn, &cs);              // v_sin_f32 / v_cos_f32
        outr[j] = m * cs;
        outi[j] = m * sn;
    }

    float* orow = out + (size_t)b * 2u * (size_t)chan_stride + (size_t)r * COLS + t * VPT;
    float* irow = orow + chan_stride;
    ((float4*)orow)[0] = make_float4(outr[0], outr[1], outr[2], outr[3]);
    ((float4*)orow)[1] = make_float4(outr[4], outr[5], outr[6], outr[7]);
    ((float4*)irow)[0] = make_float4(outi[0], outi[1], outi[2], outi[3]);
    ((float4*)irow)[1] = make_float4(outi[4], outi[5], outi[6], outi[7]);
}

extern "C" void kernel_launch(void* const* d_in, const int* in_sizes, int n_in,
                              void* d_out, int out_size, void* d_ws, size_t ws_size,
                              hipStream_t stream) {
    const float* x    = (const float*)d_in[0];
    const int*   navg = (const int*)d_in[1];
    float*       out  = (float*)d_out;

    const int R = 513;
    const int B = in_sizes[0] / (2 * R * COLS);   // 32
    const int rows = B * R;                        // 16416 blocks
    spectral_sub_kernel<<<rows, TPB, 0, stream>>>(x, navg, out, R, R * COLS);
}